// AttnDecoderRNN_26594437497554
// MI455X (gfx1250) — compile-verified
//
#include <hip/hip_runtime.h>
#include <hip/hip_bf16.h>
#include <math.h>

#define H 1024
#define V 50257
#define MAX_LEN 15

typedef float v2f __attribute__((ext_vector_type(2)));
typedef float v8f __attribute__((ext_vector_type(8)));

// ---- workspace layout (float offsets) ----
#define WS_COMB_IN   0            // 2048  [embedded | attn_applied]
#define WS_XVEC      2048         // 1024  relu(comb) output
#define WS_GI        3072         // 3072  w_ih @ x + b_ih
#define WS_GH        6144         // 3072  w_hh @ h0 + b_hh
#define WS_HNEW      9216         // 1024  new hidden
#define WS_LOGITS    10240        // V     out_W @ h_new + out_b
#define WS_LSE       60512        // 1     log-sum-exp constant

// -------------------------------------------------------------------------
// Wave-per-16-row f32 GEMV tile using V_WMMA_F32_16X16X4_F32.
//   y[r] = dot(W[r,:], x) + b[r]   (optionally ReLU)
// A fragment (16x4 f32): lane<16 holds row (lane), K={k,k+1};
//                        lane>=16 holds row (lane-16), K={k+2,k+3}.
// B fragment (4x16 f32): column-replicated x chunk -> per-lane value depends
//                        only on K, mirroring the A K-split.
// D (16x16 f32): all 16 columns identical; lane0 v8f = rows base..base+7,
//                lane16 v8f = rows base+8..base+15.
// -------------------------------------------------------------------------
template <bool RELU>
__global__ void gemv_wmma_kernel(const float* __restrict__ W,
                                 const float* __restrict__ x,
                                 const float* __restrict__ b,
                                 float* __restrict__ y,
                                 int rows, int cols)
{
    __shared__ float xs[2 * H];  // cols <= 2048
    for (int i = threadIdx.x; i < cols; i += blockDim.x) xs[i] = x[i];
    __syncthreads();

    const int lane        = threadIdx.x & 31;
    const int waveInBlk   = threadIdx.x >> 5;
    const int wavesPerBlk = blockDim.x >> 5;
    const int tile        = blockIdx.x * wavesPerBlk + waveInBlk;
    const int rowBase     = tile * 16;

    const int mrow = rowBase + (lane & 15);
    const int rowC = (mrow < rows) ? mrow : (rows - 1);   // branch-free clamp
    const int klo  = (lane >> 4) << 1;                    // 0 or 2

    const float* __restrict__ wrow = W + (size_t)rowC * (size_t)cols + klo;

    v8f acc = {};
    for (int k = 0; k < cols; k += 4) {
        v2f a = *(const v2f*)(wrow + k);        // global_load_b64, coalesced
        v2f bx;
        bx.x = xs[k + klo];
        bx.y = xs[k + klo + 1];
        acc = __builtin_amdgcn_wmma_f32_16x16x4_f32(
                  /*neg_a=*/false, a, /*neg_b=*/false, bx,
                  /*c_mod=*/(short)0, acc, /*reuse_a=*/false, /*reuse_b=*/false);
    }

    const int base = (lane == 0) ? rowBase : ((lane == 16) ? rowBase + 8 : -1);
    if (base >= 0) {
#pragma unroll
        for (int j = 0; j < 8; ++j) {
            int r = base + j;
            if (r < rows) {
                float v = acc[j] + b[r];
                if (RELU) v = fmaxf(v, 0.0f);
                y[r] = v;
            }
        }
    }
}

// -------------------------------------------------------------------------
// Stage A: embedding gather, attention scores (wave per row), softmax,
// attention-applied context, assemble comb_in. Single block, 15 waves.
// -------------------------------------------------------------------------
__global__ void stageA_kernel(const int* __restrict__ input_seq,
                              const float* __restrict__ hidden,
                              const float* __restrict__ enc,
                              const float* __restrict__ emb_W,
                              const float* __restrict__ attn_W,
                              const float* __restrict__ attn_b,
                              float* __restrict__ ws,
                              float* __restrict__ out)
{
    __shared__ float sl[MAX_LEN];
    __shared__ float saw[MAX_LEN];

    const int   tok = input_seq[0];
    const float* __restrict__ emb = emb_W + (size_t)tok * H;

    const int lane = threadIdx.x & 31;
    const int wave = threadIdx.x >> 5;

    if (wave < MAX_LEN) {
        float s = 0.0f;
        const float* __restrict__ wrow = attn_W + wave * 2 * H;
        for (int j = lane; j < 2 * H; j += 32) {
            float a = (j < H) ? emb[j] : hidden[j - H];
            s += wrow[j] * a;
        }
        for (int off = 16; off > 0; off >>= 1)
            s += __shfl_down(s, off, 32);
        if (lane == 0) sl[wave] = s + attn_b[wave];
    }
    __syncthreads();

    if (threadIdx.x == 0) {
        float m = sl[0];
        for (int l = 1; l < MAX_LEN; ++l) m = fmaxf(m, sl[l]);
        float den = 0.0f;
        for (int l = 0; l < MAX_LEN; ++l) den += expf(sl[l] - m);
        for (int l = 0; l < MAX_LEN; ++l) {
            float w = expf(sl[l] - m) / den;
            saw[l] = w;
            out[V + H + l] = w;          // output #3: attn_weights
        }
    }
    __syncthreads();

    for (int i = threadIdx.x; i < H; i += blockDim.x) {
        float e = emb[i];
        float s = 0.0f;
#pragma unroll
        for (int l = 0; l < MAX_LEN; ++l) s += saw[l] * enc[l * H + i];
        ws[WS_COMB_IN + i]     = e;      // embedded
        ws[WS_COMB_IN + H + i] = s;      // attn_applied
    }
}

// -------------------------------------------------------------------------
// GRU gate nonlinearity (PyTorch order r,z,n), writes h_new.
// -------------------------------------------------------------------------
__global__ void gru_gate_kernel(const float* __restrict__ hidden,
                                float* __restrict__ ws,
                                float* __restrict__ out)
{
    int i = blockIdx.x * blockDim.x + threadIdx.x;
    if (i >= H) return;
    float ir = ws[WS_GI + i], iz = ws[WS_GI + H + i], inn = ws[WS_GI + 2 * H + i];
    float hr = ws[WS_GH + i], hz = ws[WS_GH + H + i], hn  = ws[WS_GH + 2 * H + i];
    float r = 1.0f / (1.0f + expf(-(ir + hr)));
    float z = 1.0f / (1.0f + expf(-(iz + hz)));
    float n = tanhf(inn + r * hn);
    float h = (1.0f - z) * n + z * hidden[i];
    ws[WS_HNEW + i] = h;
    out[V + i] = h;                      // output #2: h_new
}

// -------------------------------------------------------------------------
// Single-block log-sum-exp over the V logits (200 KB, lives in L2).
// -------------------------------------------------------------------------
__global__ void lse_kernel(float* __restrict__ ws)
{
    __shared__ float red[1024];
    const int tid = threadIdx.x;

    float m = -INFINITY;
    for (int i = tid; i < V; i += blockDim.x)
        m = fmaxf(m, ws[WS_LOGITS + i]);
    red[tid] = m;
    __syncthreads();
    for (int s = blockDim.x >> 1; s > 0; s >>= 1) {
        if (tid < s) red[tid] = fmaxf(red[tid], red[tid + s]);
        __syncthreads();
    }
    const float gmax = red[0];
    __syncthreads();

    float sum = 0.0f;
    for (int i = tid; i < V; i += blockDim.x)
        sum += expf(ws[WS_LOGITS + i] - gmax);
    red[tid] = sum;
    __syncthreads();
    for (int s = blockDim.x >> 1; s > 0; s >>= 1) {
        if (tid < s) red[tid] += red[tid + s];
        __syncthreads();
    }
    if (tid == 0) ws[WS_LSE] = gmax + logf(red[0]);
}

__global__ void finalize_kernel(const float* __restrict__ ws,
                                float* __restrict__ out)
{
    int i = blockIdx.x * blockDim.x + threadIdx.x;
    if (i < V) out[i] = ws[WS_LOGITS + i] - ws[WS_LSE];
}

// -------------------------------------------------------------------------
extern "C" void kernel_launch(void* const* d_in, const int* in_sizes, int n_in,
                              void* d_out, int out_size, void* d_ws, size_t ws_size,
                              hipStream_t stream)
{
    (void)in_sizes; (void)n_in; (void)out_size; (void)ws_size;

    const int*   input_seq = (const int*)  d_in[0];
    const float* hidden    = (const float*)d_in[1];
    const float* enc       = (const float*)d_in[2];
    const float* emb_W     = (const float*)d_in[3];
    const float* attn_W    = (const float*)d_in[4];
    const float* attn_b    = (const float*)d_in[5];
    const float* comb_W    = (const float*)d_in[6];
    const float* comb_b    = (const float*)d_in[7];
    const float* w_ih      = (const float*)d_in[8];
    const float* w_hh      = (const float*)d_in[9];
    const float* b_ih      = (const float*)d_in[10];
    const float* b_hh      = (const float*)d_in[11];
    const float* out_W     = (const float*)d_in[12];
    const float* out_b     = (const float*)d_in[13];

    float* out = (float*)d_out;
    float* ws  = (float*)d_ws;

    // 1) embedding + attention + context  (tiny, single block of 15 waves)
    stageA_kernel<<<1, 480, 0, stream>>>(input_seq, hidden, enc,
                                         emb_W, attn_W, attn_b, ws, out);

    // 2) x = relu(comb_W @ comb_in + comb_b)   rows=1024 cols=2048 -> 64 tiles
    gemv_wmma_kernel<true><<<8, 256, 0, stream>>>(
        comb_W, ws + WS_COMB_IN, comb_b, ws + WS_XVEC, H, 2 * H);

    // 3) GRU gate GEMVs: rows=3072 cols=1024 -> 192 tiles each
    gemv_wmma_kernel<false><<<24, 256, 0, stream>>>(
        w_ih, ws + WS_XVEC, b_ih, ws + WS_GI, 3 * H, H);
    gemv_wmma_kernel<false><<<24, 256, 0, stream>>>(
        w_hh, hidden, b_hh, ws + WS_GH, 3 * H, H);

    gru_gate_kernel<<<(H + 255) / 256, 256, 0, stream>>>(hidden, ws, out);

    // 4) logits = out_W @ h_new + out_b   rows=50257 -> 3142 tiles, 393 blocks
    {
        int tiles  = (V + 15) / 16;
        int blocks = (tiles + 7) / 8;
        gemv_wmma_kernel<false><<<blocks, 256, 0, stream>>>(
            out_W, ws + WS_HNEW, out_b, ws + WS_LOGITS, V, H);
    }

    // 5) log_softmax
    lse_kernel<<<1, 1024, 0, stream>>>(ws);
    finalize_kernel<<<(V + 255) / 256, 256, 0, stream>>>(ws, out);
}